// Evaluate_14963666059433
// MI455X (gfx1250) — compile-verified
//
#include <hip/hip_runtime.h>
#include <stdint.h>

// Problem constants (from reference setup_inputs)
#define BB   8
#define CC   256
#define NUM  24
#define HH   64
#define WW   64
#define HWP  4096          // HH*WW
#define TOPK 8             // NUM / FILTER_SIZE
#define INV_TEMP 100.0f    // 1 / 0.01

// ---------------------------------------------------------------------------
// 32x32 tiled transpose, [C x Q] -> [Q x C], blockIdx.z selects matrix.
// Pure CDNA5 data-mover kernel: the tile payload never enters VGPRs.
//   global -> LDS : global_load_async_to_lds_b128   (ASYNCcnt)
//   LDS -> global : global_store_async_from_lds_b32 (ASYNCcnt)
// Row stride 36 floats keeps each 4-column quad 16B-aligned for B128 and the
// outstanding async stores drain at s_endpgm's implicit wait-idle.
// ---------------------------------------------------------------------------
__global__ __launch_bounds__(256) void transpose_async_kernel(
    const float* __restrict__ in, float* __restrict__ out, int C, int Q)
{
    __shared__ float tile[32][36];
    const size_t mat = (size_t)blockIdx.z * (size_t)C * (size_t)Q;
    const int qb = blockIdx.x * 32;
    const int cb = blockIdx.y * 32;
    const int t  = threadIdx.x;               // 0..255

    {   // load phase: thread t copies one 16B quad of row (cb+ci)
        const int ci = t >> 3;                // 0..31
        const int qq = (t & 7) << 2;          // 0,4,...,28
        const float* gp = in + mat + (size_t)(cb + ci) * (size_t)Q + (size_t)(qb + qq);
        const unsigned lp = (unsigned)(uintptr_t)(&tile[ci][qq]);    // low 32b = LDS offset
        const unsigned long long ga = (unsigned long long)(uintptr_t)gp;
        asm volatile("global_load_async_to_lds_b128 %0, %1, off"
                     :: "v"(lp), "v"(ga) : "memory");
    }
    asm volatile("s_wait_asynccnt 0x0" ::: "memory");
    __syncthreads();

    {   // store phase: transposed, coalesced along channels
        const int cl = t & 31;                // channel within tile
        const int q0 = t >> 5;                // 0..7
#pragma unroll
        for (int i = 0; i < 4; ++i) {
            const int qi = q0 + 8 * i;
            const float* gp = out + mat + (size_t)(qb + qi) * (size_t)C + (size_t)(cb + cl);
            const unsigned lp = (unsigned)(uintptr_t)(&tile[cl][qi]);
            const unsigned long long ga = (unsigned long long)(uintptr_t)gp;
            asm volatile("global_store_async_from_lds_b32 %0, %1, off"
                         :: "v"(ga), "v"(lp) : "memory");
        }
    }
}

// ---------------------------------------------------------------------------
// Main kernel: one wave32 per (batch, pixel).
//   TRANS=true : Lp = LT [B][HW][C], Rp = RT [B*HW][C]   (contiguous rows)
//   TRANS=false: Lp = L  [B][C][HW], Rp = R  [C][B*HW]   (strided fallback)
// ---------------------------------------------------------------------------
template <bool TRANS>
__global__ __launch_bounds__(256) void corr_topk_kernel(
    const float* __restrict__ Lp,
    const float* __restrict__ Rp,
    const int*   __restrict__ offx,   // [B][NUM][HW]
    const int*   __restrict__ offy,   // [B][NUM][HW]
    float*       __restrict__ out)    // off_x | off_y | corr, each [B][TOPK][HW]
{
    const int lane = threadIdx.x & 31;
    const int wave = threadIdx.x >> 5;
    const int pid  = blockIdx.x * 8 + wave;   // global pixel id, 0..B*HW-1
    const int b    = pid >> 12;               // / HWP
    const int p    = pid & (HWP - 1);

    // Lanes 0..23 hold candidate `lane`'s target index.
    int myind = 0;
    if (lane < NUM) {
        const int o  = (b * NUM + lane) * HWP + p;
        const int ox = offx[o];
        const int oy = offy[o];
        myind = (p / WW + ox) * WW + (p % WW) + oy;   // in [0, HWP)
    }

    // Left feature row: lane owns channels 8*lane .. 8*lane+7.
    float lv[8];
    if (TRANS) {
        const float4* lr = (const float4*)(Lp + (size_t)pid * CC + 8 * lane);
        const float4 a = lr[0], c4 = lr[1];
        lv[0]=a.x; lv[1]=a.y; lv[2]=a.z; lv[3]=a.w;
        lv[4]=c4.x; lv[5]=c4.y; lv[6]=c4.z; lv[7]=c4.w;
    } else {
#pragma unroll
        for (int k = 0; k < 8; ++k)
            lv[k] = Lp[((size_t)b * CC + (8 * lane + k)) * HWP + p];
    }

    // 24 gathered row dots; fully unrolled so the 48 b128 loads provide MLP.
    float acc[NUM];
#pragma unroll
    for (int n = 0; n < NUM; ++n) {
        const int ind = __shfl(myind, n, 32);
        const size_t g = (size_t)b * HWP + (size_t)ind;
        float rv[8];
        if (TRANS) {
            const float4* rr = (const float4*)(Rp + g * CC + 8 * lane); // coalesced row
            const float4 a = rr[0], c4 = rr[1];
            rv[0]=a.x; rv[1]=a.y; rv[2]=a.z; rv[3]=a.w;
            rv[4]=c4.x; rv[5]=c4.y; rv[6]=c4.z; rv[7]=c4.w;
        } else {
#pragma unroll
            for (int k = 0; k < 8; ++k)
                rv[k] = Rp[(size_t)(8 * lane + k) * ((size_t)BB * HWP) + g];
        }
        float a = 0.f;
#pragma unroll
        for (int k = 0; k < 8; ++k) a = fmaf(lv[k], rv[k], a);
        acc[n] = a;
    }

    // Wave-wide reduction per candidate (wave32 xor butterfly); result in all lanes.
#pragma unroll
    for (int n = 0; n < NUM; ++n) {
        float a = acc[n];
#pragma unroll
        for (int m = 16; m >= 1; m >>= 1) a += __shfl_xor(a, m, 32);
        acc[n] = a * INV_TEMP;
    }

    // Softmax over 24 candidates (redundant across lanes; registers only).
    float mx = acc[0];
#pragma unroll
    for (int n = 1; n < NUM; ++n) mx = fmaxf(mx, acc[n]);
    float s = 0.f;
#pragma unroll
    for (int n = 0; n < NUM; ++n) { acc[n] = expf(acc[n] - mx); s += acc[n]; }
    const float inv = 1.0f / s;
#pragma unroll
    for (int n = 0; n < NUM; ++n) acc[n] *= inv;

    // Top-8 by repeated argmax; strict '>' + ascending scan => lowest index on
    // ties, descending output order (matches jax.lax.top_k). Lane j keeps rank j.
    unsigned used = 0u;
    float myv = 0.f;
    int   myi = 0;
    for (int j = 0; j < TOPK; ++j) {
        float bv = -1.f; int bi = 0;
#pragma unroll
        for (int n = 0; n < NUM; ++n) {
            const bool ok = (((used >> n) & 1u) == 0u) && (acc[n] > bv);
            bv = ok ? acc[n] : bv;
            bi = ok ? n : bi;
        }
        used |= (1u << bi);
        if (j == lane) { myv = bv; myi = bi; }
    }

    const int tind = __shfl(myind, myi, 32);  // per-lane src lane (bpermute)
    if (lane < TOPK) {
        const size_t plane = (size_t)BB * TOPK * HWP;
        const size_t o = (size_t)(b * TOPK + lane) * HWP + p;
        out[o]             = (float)(tind / WW - p / WW);
        out[plane + o]     = (float)(tind % WW - p % WW);
        out[2 * plane + o] = myv;
    }
}

// ---------------------------------------------------------------------------
extern "C" void kernel_launch(void* const* d_in, const int* in_sizes, int n_in,
                              void* d_out, int out_size, void* d_ws, size_t ws_size,
                              hipStream_t stream)
{
    (void)in_sizes; (void)n_in; (void)out_size;
    const float* L  = (const float*)d_in[0];   // [B][C][HW]
    const float* R  = (const float*)d_in[1];   // [C][B*HW]
    const int*   ox = (const int*)d_in[2];     // [B][NUM][HW]
    const int*   oy = (const int*)d_in[3];     // [B][NUM][HW]
    float* out = (float*)d_out;

    const size_t elems = (size_t)BB * HWP * CC;          // 8.39M floats each
    const size_t need  = 2 * elems * sizeof(float);      // 64 MB workspace

    if (ws_size >= need) {
        float* RT = (float*)d_ws;        // [B*HW][C]
        float* LT = RT + elems;          // [B][HW][C]
        transpose_async_kernel<<<dim3((BB * HWP) / 32, CC / 32, 1), 256, 0, stream>>>(
            R, RT, CC, BB * HWP);
        transpose_async_kernel<<<dim3(HWP / 32, CC / 32, BB), 256, 0, stream>>>(
            L, LT, CC, HWP);
        corr_topk_kernel<true><<<(BB * HWP) / 8, 256, 0, stream>>>(LT, RT, ox, oy, out);
    } else {
        corr_topk_kernel<false><<<(BB * HWP) / 8, 256, 0, stream>>>(L, R, ox, oy, out);
    }
}